// GNNModule_11304353923222
// MI455X (gfx1250) — compile-verified
//
#include <hip/hip_runtime.h>
#include <math.h>

// MI455X / gfx1250, wave32.
// eg[b,i] = sum_j adj * (H^T H) computed via V_WMMA_F32_16X16X4_F32 with
// SYMMETRY: only jt>=it tiles are computed; each tile C is masked twice
// (adj[i,j].C into eg_i, adj[j,i].C^T into eg_j).  G never hits memory.
// adj@u also via WMMA (N padded 8->16), deg folded into that pass.

typedef float v2f __attribute__((ext_vector_type(2)));
typedef float v8f __attribute__((ext_vector_type(8)));

#define NN    1024
#define BB    16
#define NU    8
#define NH    12
#define NMLP  16

__device__ __forceinline__ float sigmoidf(float x) { return 1.0f / (1.0f + __expf(-x)); }

// ---------------------------------------------------------------------------
// yh[b,j] = sum_i y[b,i]*H[b,i,j];  hh[b,j] = sum_i H[b,i,j]^2
__global__ __launch_bounds__(256) void k_colstats(
    const float* __restrict__ y, const float* __restrict__ H,
    float* __restrict__ yh, float* __restrict__ hh)
{
  const int b = blockIdx.x;
  const int j = (blockIdx.y << 8) + threadIdx.x;
  const float* Hb = H + (size_t)b * NN * NN + j;
  const float* yb = y + b * NN;
  float ya = 0.0f, ha = 0.0f;
#pragma unroll 4
  for (int i = 0; i < NN; ++i) {
    const float h = Hb[(size_t)i * NN];
    ya = fmaf(yb[i], h, ya);
    ha = fmaf(h, h, ha);
  }
  yh[b * NN + j] = ya;
  hh[b * NN + j] = ha;
}

// ---------------------------------------------------------------------------
__global__ __launch_bounds__(256) void k_init_u(
    const float* __restrict__ yh, const float* __restrict__ hh,
    const float* __restrict__ W1_w, const float* __restrict__ W1_b,
    const float* __restrict__ b1, float* __restrict__ u)
{
  const int node = blockIdx.x * 256 + threadIdx.x;      // 0..16383
  const float f0 = yh[node], f1 = hh[node];
#pragma unroll
  for (int o = 0; o < NU; ++o)
    u[node * NU + o] = fmaf(W1_w[2 * o], f0, fmaf(W1_w[2 * o + 1], f1, W1_b[o] + b1[o]));
}

// ---------------------------------------------------------------------------
__global__ __launch_bounds__(256) void k_zero(float* __restrict__ p)
{
  p[blockIdx.x * 256 + threadIdx.x] = 0.0f;
}

// ---------------------------------------------------------------------------
// Symmetric masked Gram reduction.
// Block = (b, g).  Phase 0: i-tile it=g (64-g j-tiles); phase 1: it=63-g
// (g+1 j-tiles) -> 65 tiles per block, uniform.  Per phase, K split in two
// 512 chunks; A-panel H[kchunk, i0:i0+16] staged in LDS ([i][k], pad 2 ->
// conflict-free 8B-aligned ds_load_2addr_b64).  Waves own j-tiles jt=it+w+8q.
// Epilogue per tile: i-side into per-lane regs (LDS-reduced per phase, then
// one global atomic per row); j-side (jt>it) masks C^T with the transposed
// adj tile (8 consecutive floats/lane -> b128) and atomics into eg[b,j].
__global__ __launch_bounds__(256) void k_eg(
    const float* __restrict__ H, const float* __restrict__ adj,
    float* __restrict__ eg)
{
  __shared__ float Ash[16 * 514];          // 32.1 KB
  __shared__ float red[16];
  const int b    = blockIdx.x;
  const int g    = blockIdx.y;             // 0..31
  const int tid  = threadIdx.x;
  const int wave = tid >> 5;
  const int lane = tid & 31;
  const int lh   = lane & 15;
  const int hi   = lane >> 4;              // lane half: 0 or 1

  const float* Hb = H   + (size_t)b * NN * NN;
  const float* Ab = adj + (size_t)b * NN * NN;

  for (int phase = 0; phase < 2; ++phase) {
    const int it = phase ? (63 - g) : g;
    const int i0 = it << 4;

    float egp[8];
#pragma unroll
    for (int v = 0; v < 8; ++v) egp[v] = 0.0f;

    for (int kc = 0; kc < 2; ++kc) {
      const int kbase = kc * 512;
      __syncthreads();                     // guard prior Ash / red readers
      for (int e = tid; e < 512 * 16; e += 256) {
        const int i = e & 15, k = e >> 4;  // i fastest -> coalesced 64B rows
        Ash[i * 514 + k] = Hb[(size_t)(kbase + k) * NN + (i0 + i)];
      }
      __syncthreads();

      for (int jt = it + wave; jt < 64; jt += 8) {
        const int j0 = jt << 4;
        v8f c = {};
        const float* Bcol = Hb + (size_t)kbase * NN + j0 + lh;
        const float* Arow = &Ash[lh * 514 + 2 * hi];
#pragma unroll 4
        for (int kk = 0; kk < 512; kk += 4) {
          // A 16x4: lane M=lh, V0/V1 = K {kk,kk+1} (lanes<16) / {kk+2,kk+3}
          v2f a = *(const v2f*)(Arow + kk);
          // B 4x16: V0 row K=kk+2*hi, V1 row K=kk+2*hi+1, N = lh
          const float* Bp = Bcol + (size_t)(kk + 2 * hi) * NN;
          v2f bf; bf[0] = Bp[0]; bf[1] = Bp[NN];
          c = __builtin_amdgcn_wmma_f32_16x16x4_f32(false, a, false, bf,
                                                    (short)0, c, false, false);
        }
        // C[v] = partial G(i0+v+8*hi, j0+lh) over this k-chunk.
        // i-side: adj[i,j] . C  ->  eg rows of this i-tile
        const float* At = Ab + (size_t)(i0 + 8 * hi) * NN + j0 + lh;
#pragma unroll
        for (int v = 0; v < 8; ++v)
          egp[v] = fmaf(At[(size_t)v * NN], c[v], egp[v]);
        // j-side: adj[j,i] . C^T  ->  eg rows of the j-tile (skip diagonal)
        if (jt > it) {
          const float* AtT = Ab + (size_t)(j0 + lh) * NN + i0 + 8 * hi;
          float colp = 0.0f;
#pragma unroll
          for (int v = 0; v < 8; ++v)       // 8 consecutive floats -> b128
            colp = fmaf(AtT[v], c[v], colp);
          atomicAdd(&eg[b * NN + j0 + lh], colp);
        }
      }
    }

    // phase reduce of the i-side
    __syncthreads();
    if (tid < 16) red[tid] = 0.0f;
    __syncthreads();
#pragma unroll
    for (int v = 0; v < 8; ++v) atomicAdd(&red[8 * hi + v], egp[v]);
    __syncthreads();
    if (tid < 16) atomicAdd(&eg[b * NN + i0 + tid], red[tid]);
  }
}

// ---------------------------------------------------------------------------
// mv[b,i,d] = sum_j adj[b,i,j]*u[b,j,d]  (d<8, N padded to 16)  +  deg[b,i].
// Block = (b, 16-row i-tile); wave w owns K range [128w,128w+128); LDS reduce.
__global__ __launch_bounds__(256) void k_mv(
    const float* __restrict__ adj, const float* __restrict__ u,
    float* __restrict__ mv, float* __restrict__ deg)
{
  __shared__ float mvsh[16 * NU];
  __shared__ float degsh[16];
  const int b    = blockIdx.x;
  const int i0   = blockIdx.y << 4;
  const int tid  = threadIdx.x;
  const int wave = tid >> 5, lane = tid & 31;
  const int lh   = lane & 15, hi = lane >> 4;
  if (tid < 16 * NU) mvsh[tid] = 0.0f;
  if (tid < 16) degsh[tid] = 0.0f;
  __syncthreads();

  const float* Ab = adj + (size_t)b * NN * NN;
  const float* Ub = u   + (size_t)b * NN * NU;
  const int k0 = wave << 7;

  v8f c = {};
  float dp = 0.0f;
  const float* Arow = Ab + (size_t)(i0 + lh) * NN + k0 + 2 * hi;
#pragma unroll 4
  for (int kk = 0; kk < 128; kk += 4) {
    v2f a = *(const v2f*)(Arow + kk);          // contiguous K pair, 8B aligned
    dp += a[0] + a[1];                          // row sum -> deg
    const int krow = k0 + kk + 2 * hi;
    v2f bf;
    bf[0] = (lh < NU) ? Ub[(size_t)krow * NU + lh]       : 0.0f;
    bf[1] = (lh < NU) ? Ub[(size_t)(krow + 1) * NU + lh] : 0.0f;
    c = __builtin_amdgcn_wmma_f32_16x16x4_f32(false, a, false, bf,
                                              (short)0, c, false, false);
  }
  atomicAdd(&degsh[lh], dp);
  if (lh < NU) {
#pragma unroll
    for (int v = 0; v < 8; ++v)
      atomicAdd(&mvsh[(v + 8 * hi) * NU + lh], c[v]);
  }
  __syncthreads();
  if (tid < 16 * NU)
    mv[((size_t)b * NN + i0 + (tid >> 3)) * NU + (tid & 7)] = mvsh[tid];
  if (tid < 16)
    deg[b * NN + i0 + tid] = degsh[tid];
}

// ---------------------------------------------------------------------------
// Per-node msg-MLP + GRU + (last) readout/softmax. One thread per node.
__global__ __launch_bounds__(256) void k_node(
    const float* __restrict__ mv, const float* __restrict__ deg,
    const float* __restrict__ eg, const float* __restrict__ r,
    const float* __restrict__ nu_r,
    const float* __restrict__ Wm1_w, const float* __restrict__ Wm1_b,
    const float* __restrict__ Wm2_w, const float* __restrict__ Wm2_b,
    const float* __restrict__ w_ih, const float* __restrict__ w_hh,
    const float* __restrict__ b_ih, const float* __restrict__ b_hh,
    const float* __restrict__ W2_w, const float* __restrict__ W2_b,
    const float* __restrict__ b2,
    const float* __restrict__ Wr1_w, const float* __restrict__ Wr1_b,
    const float* __restrict__ Wr2_w, const float* __restrict__ Wr2_b,
    float* __restrict__ u, float* __restrict__ s, float* __restrict__ out,
    int first, int last)
{
  const int node = blockIdx.x * 256 + threadIdx.x;

  float agg[3 * NU];
  const float dv = deg[node], ev = eg[node];
#pragma unroll
  for (int o = 0; o < NU; ++o) {
    agg[o]          = u[node * NU + o] * dv;
    agg[NU + o]     = mv[node * NU + o];
    agg[2 * NU + o] = ev;
  }
  float h1[NMLP];
#pragma unroll
  for (int t = 0; t < NMLP; ++t) {
    float a = Wm1_b[t];
#pragma unroll
    for (int c = 0; c < 3 * NU; ++c) a = fmaf(Wm1_w[t * 3 * NU + c], agg[c], a);
    h1[t] = fmaxf(a, 0.0f);
  }
  float x[NU + 2];
#pragma unroll
  for (int o = 0; o < NU; ++o) {
    float a = Wm2_b[o];
#pragma unroll
    for (int t = 0; t < NMLP; ++t) a = fmaf(Wm2_w[o * NMLP + t], h1[t], a);
    x[o] = a;
  }
  x[NU]     = r[node];
  x[NU + 1] = fmaxf(nu_r[node], 1e-10f);

  float sv[NH];
#pragma unroll
  for (int g = 0; g < NH; ++g) sv[g] = first ? 0.0f : s[node * NH + g];

  float gx[3 * NH], gh[3 * NH];
#pragma unroll
  for (int g = 0; g < 3 * NH; ++g) {
    float a = b_ih[g];
#pragma unroll
    for (int c = 0; c < NU + 2; ++c) a = fmaf(w_ih[g * (NU + 2) + c], x[c], a);
    gx[g] = a;
    float h = b_hh[g];
#pragma unroll
    for (int c = 0; c < NH; ++c) h = fmaf(w_hh[g * NH + c], sv[c], h);
    gh[g] = h;
  }
#pragma unroll
  for (int g = 0; g < NH; ++g) {
    const float rg = sigmoidf(gx[g] + gh[g]);
    const float zg = sigmoidf(gx[NH + g] + gh[NH + g]);
    const float ng = tanhf(gx[2 * NH + g] + rg * gh[2 * NH + g]);
    sv[g] = (1.0f - zg) * ng + zg * sv[g];
  }
#pragma unroll
  for (int g = 0; g < NH; ++g) s[node * NH + g] = sv[g];

  float un[NU];
#pragma unroll
  for (int o = 0; o < NU; ++o) {
    float a = W2_b[o] + b2[o];
#pragma unroll
    for (int c = 0; c < NH; ++c) a = fmaf(W2_w[o * NH + c], sv[c], a);
    un[o] = a;
    u[node * NU + o] = a;
  }

  if (last) {
    float h2[NMLP];
#pragma unroll
    for (int t = 0; t < NMLP; ++t) {
      float a = Wr1_b[t];
#pragma unroll
      for (int c = 0; c < NU; ++c) a = fmaf(Wr1_w[t * NU + c], un[c], a);
      h2[t] = fmaxf(a, 0.0f);
    }
    float l0 = Wr2_b[0], l1 = Wr2_b[1];
#pragma unroll
    for (int t = 0; t < NMLP; ++t) {
      l0 = fmaf(Wr2_w[t],        h2[t], l0);
      l1 = fmaf(Wr2_w[NMLP + t], h2[t], l1);
    }
    const float mx = fmaxf(l0, l1);
    const float e0 = __expf(l0 - mx), e1 = __expf(l1 - mx);
    const float inv = 1.0f / (e0 + e1);
    const float p0 = e0 * inv, p1 = e1 * inv;
    const float Q0 = -0.70710678118654752f, Q1 = 0.70710678118654752f;
    const float xh = p0 * Q0 + p1 * Q1;
    float nu = p0 * (Q0 - xh) * (Q0 - xh) + p1 * (Q1 - xh) * (Q1 - xh);
    out[node]           = xh;
    out[BB * NN + node] = fmaxf(nu, 1e-10f);
  }
}

// ---------------------------------------------------------------------------
extern "C" void kernel_launch(void* const* d_in, const int* in_sizes, int n_in,
                              void* d_out, int out_size, void* d_ws, size_t ws_size,
                              hipStream_t stream) {
  (void)in_sizes; (void)n_in; (void)out_size; (void)ws_size;
  const float* y    = (const float*)d_in[0];
  const float* H    = (const float*)d_in[1];
  const float* r    = (const float*)d_in[2];
  const float* nu_r = (const float*)d_in[3];
  const float* adj  = (const float*)d_in[4];
  const float* W1_w = (const float*)d_in[5];
  const float* W1_b = (const float*)d_in[6];
  const float* b1   = (const float*)d_in[7];
  const float* Wm1_w= (const float*)d_in[8];
  const float* Wm1_b= (const float*)d_in[9];
  const float* Wm2_w= (const float*)d_in[10];
  const float* Wm2_b= (const float*)d_in[11];
  const float* w_ih = (const float*)d_in[12];
  const float* w_hh = (const float*)d_in[13];
  const float* b_ih = (const float*)d_in[14];
  const float* b_hh = (const float*)d_in[15];
  const float* W2_w = (const float*)d_in[16];
  const float* W2_b = (const float*)d_in[17];
  const float* b2   = (const float*)d_in[18];
  const float* Wr1_w= (const float*)d_in[19];
  const float* Wr1_b= (const float*)d_in[20];
  const float* Wr2_w= (const float*)d_in[21];
  const float* Wr2_b= (const float*)d_in[22];

  float* ws  = (float*)d_ws;                 // 2 MB of fp32 scratch used
  float* yh  = ws;                           // 16384
  float* hh  = ws + 16384;                   // 16384
  float* eg  = ws + 32768;                   // 16384 (atomically accumulated)
  float* deg = ws + 49152;                   // 16384
  float* u   = ws + 65536;                   // 131072
  float* mv  = ws + 196608;                  // 131072
  float* s   = ws + 327680;                  // 196608

  k_colstats<<<dim3(BB, 4), 256, 0, stream>>>(y, H, yh, hh);
  k_init_u  <<<64, 256, 0, stream>>>(yh, hh, W1_w, W1_b, b1, u);
  k_zero    <<<64, 256, 0, stream>>>(eg);    // eg is atomic-accumulated: re-zero every call
  k_eg      <<<dim3(BB, 32), 256, 0, stream>>>(H, adj, eg);
  for (int it = 0; it < 2; ++it) {
    k_mv  <<<dim3(BB, 64), 256, 0, stream>>>(adj, u, mv, deg);
    k_node<<<64, 256, 0, stream>>>(mv, deg, eg, r, nu_r,
        Wm1_w, Wm1_b, Wm2_w, Wm2_b, w_ih, w_hh, b_ih, b_hh,
        W2_w, W2_b, b2, Wr1_w, Wr1_b, Wr2_w, Wr2_b,
        u, s, (float*)d_out, it == 0 ? 1 : 0, it == 1 ? 1 : 0);
  }
}